// BitLinear_52707838657223
// MI455X (gfx1250) — compile-verified
//
#include <hip/hip_runtime.h>

typedef float v2f __attribute__((ext_vector_type(2)));
typedef float v8f __attribute__((ext_vector_type(8)));

#define BM 128
#define BN 128
#define BK 16
#define LDK 20   // padded LDS row stride in dwords (multiple of 4 -> float4-aligned)

// ---------------- Pass 1: per-row absmax scale ----------------
__global__ __launch_bounds__(256)
void bitlinear_scale_kernel(const float* __restrict__ w, float* __restrict__ scale, int K) {
  __shared__ float red[256];
  const int row = blockIdx.x;
  const float* wr = w + (size_t)row * K;
  float m = 0.0f;
  for (int i = threadIdx.x; i < K; i += 256) m = fmaxf(m, fabsf(wr[i]));
  red[threadIdx.x] = m;
  __syncthreads();
  for (int s = 128; s > 0; s >>= 1) {
    if ((int)threadIdx.x < s) red[threadIdx.x] = fmaxf(red[threadIdx.x], red[threadIdx.x + s]);
    __syncthreads();
  }
  if (threadIdx.x == 0) scale[row] = fmaxf(red[0], 1e-6f);
}

// CDNA5 async global->LDS copy (tracked by ASYNCcnt); per-lane 16B transfer.
__device__ __forceinline__ void async_copy_b128(unsigned lds_off, const float* gptr) {
  asm volatile("global_load_async_to_lds_b128 %0, %1, off"
               :: "v"(lds_off), "v"(gptr)
               : "memory");
}
__device__ __forceinline__ void wait_async0() {
  asm volatile("s_wait_asynccnt 0x0" ::: "memory");
}

// ---------------- Pass 2: fp32 WMMA GEMM with on-the-fly ternarize ----------------
__global__ __launch_bounds__(256)
void bitlinear_gemm_kernel(const float* __restrict__ X, const float* __restrict__ W,
                           const float* __restrict__ scale, const float* __restrict__ bias,
                           float* __restrict__ out, int M, int N, int K) {
  __shared__ float ldsA[2][BM * LDK];  // x tile   [row-in-M][k]  (filled by async DMA)
  __shared__ float ldsB[2][BM * LDK];  // qw tile  [row-in-N][k]  (reg path + ternarize)

  const int tid    = threadIdx.x;
  const int lane   = tid & 31;
  const int wave   = tid >> 5;        // 8 waves
  const int wm     = wave & 3;        // 4 waves along M (32 rows each)
  const int wn     = wave >> 2;       // 2 waves along N (64 cols each)
  const int lane16 = lane & 15;
  const int lhi    = lane >> 4;       // 0: K=0,1   1: K=2,3 (f32 WMMA A/B layout)

  const int rowBase = blockIdx.x * BM;   // M tile
  const int colBase = blockIdx.y * BN;   // N tile (output features)

  // global<->LDS loader mapping: 2x 16B per thread per matrix
  const int r0 = tid >> 2;           // rows 0..63
  const int r1 = r0 + 64;            // rows 64..127
  const int c0 = (tid & 3) * 4;      // k offset {0,4,8,12}

  const float s0 = scale[colBase + r0];
  const float s1 = scale[colBase + r1];

  const float* xg0 = X + (size_t)(rowBase + r0) * K + c0;
  const float* xg1 = X + (size_t)(rowBase + r1) * K + c0;
  const float* wg0 = W + (size_t)(colBase + r0) * K + c0;
  const float* wg1 = W + (size_t)(colBase + r1) * K + c0;

  // LDS byte offsets for the async DMA destinations (flat->LDS truncates to addr[31:0])
  const unsigned ldsA0    = (unsigned)(uintptr_t)&ldsA[0][r0 * LDK + c0];
  const unsigned ldsA1    = (unsigned)(uintptr_t)&ldsA[0][r1 * LDK + c0];
  const unsigned bufBytes = (unsigned)(BM * LDK * sizeof(float));

  float4 wr0, wr1;

  auto tern = [](float4 v, float s) -> float4 {
    const float t = 0.5f * s;
    float4 r;
    r.x = (fabsf(v.x) > t) ? copysignf(s, v.x) : 0.0f;
    r.y = (fabsf(v.y) > t) ? copysignf(s, v.y) : 0.0f;
    r.z = (fabsf(v.z) > t) ? copysignf(s, v.z) : 0.0f;
    r.w = (fabsf(v.w) > t) ? copysignf(s, v.w) : 0.0f;
    return r;
  };

  auto wstore = [&](int buf) {
    *(float4*)&ldsB[buf][r0 * LDK + c0] = tern(wr0, s0);
    *(float4*)&ldsB[buf][r1 * LDK + c0] = tern(wr1, s1);
  };

  v8f acc[2][4] = {};

  // prologue: tile 0
  async_copy_b128(ldsA0, xg0);
  async_copy_b128(ldsA1, xg1);
  wr0 = *(const float4*)wg0;
  wr1 = *(const float4*)wg1;
  wstore(0);
  wait_async0();

  const int NK = K / BK;
  for (int kt = 0; kt < NK; ++kt) {
    __syncthreads();                    // tile kt resident in buf; prior readers done
    const int buf = kt & 1;

    if (kt + 1 < NK) {                  // start DMA + w loads for tile kt+1 into buf^1
      const size_t off = (size_t)(kt + 1) * BK;
      const unsigned b = (buf ^ 1) ? bufBytes : 0u;
      async_copy_b128(ldsA0 + b, xg0 + off);
      async_copy_b128(ldsA1 + b, xg1 + off);
      wr0 = *(const float4*)(wg0 + off);
      wr1 = *(const float4*)(wg1 + off);
    }

    const float* xs = ldsA[buf];
    const float* ws = ldsB[buf];

#pragma unroll
    for (int k4 = 0; k4 < BK / 4; ++k4) {
      v2f a[2], b[4];
#pragma unroll
      for (int mt = 0; mt < 2; ++mt)
        a[mt] = *(const v2f*)(xs + (wm * 32 + mt * 16 + lane16) * LDK + k4 * 4 + lhi * 2);
#pragma unroll
      for (int nt = 0; nt < 4; ++nt)
        b[nt] = *(const v2f*)(ws + (wn * 64 + nt * 16 + lane16) * LDK + k4 * 4 + lhi * 2);
#pragma unroll
      for (int mt = 0; mt < 2; ++mt)
#pragma unroll
        for (int nt = 0; nt < 4; ++nt)
          acc[mt][nt] = __builtin_amdgcn_wmma_f32_16x16x4_f32(
              false, a[mt], false, b[nt], (short)0, acc[mt][nt], false, false);
    }

    if (kt + 1 < NK) wstore(buf ^ 1);
    wait_async0();                      // my DMA done; barrier at loop top publishes it
  }

  // epilogue: C/D layout -> VGPR j holds M=j (lanes 0-15) / M=j+8 (lanes 16-31), N=lane16
  float bn[4];
#pragma unroll
  for (int nt = 0; nt < 4; ++nt) bn[nt] = bias[colBase + wn * 64 + nt * 16 + lane16];

#pragma unroll
  for (int mt = 0; mt < 2; ++mt) {
#pragma unroll
    for (int j = 0; j < 8; ++j) {
      const int row = rowBase + wm * 32 + mt * 16 + j + lhi * 8;
#pragma unroll
      for (int nt = 0; nt < 4; ++nt) {
        const int col = colBase + wn * 64 + nt * 16 + lane16;
        out[(size_t)row * N + col] = acc[mt][nt][j] + bn[nt];
      }
    }
  }
}

extern "C" void kernel_launch(void* const* d_in, const int* in_sizes, int n_in,
                              void* d_out, int out_size, void* d_ws, size_t ws_size,
                              hipStream_t stream) {
  const float* x    = (const float*)d_in[0];
  const float* w    = (const float*)d_in[1];
  const float* bias = (const float*)d_in[2];
  float* out = (float*)d_out;

  const int Dout = in_sizes[2];              // 2048
  const int Din  = in_sizes[1] / Dout;       // 2048
  const int M    = in_sizes[0] / Din;        // 16384

  float* scale = (float*)d_ws;               // Dout floats of scratch

  bitlinear_scale_kernel<<<Dout, 256, 0, stream>>>(w, scale, Din);

  dim3 grid(M / BM, Dout / BN);
  bitlinear_gemm_kernel<<<grid, 256, 0, stream>>>(x, w, scale, bias, out, M, Dout, Din);
}